// TestCosine_28467043237895
// MI455X (gfx1250) — compile-verified
//
#include <hip/hip_runtime.h>
#include <math.h>

typedef __attribute__((ext_vector_type(2))) float v2f;
typedef __attribute__((ext_vector_type(8))) float v8f;

#define NROWS 8192
#define MROWS 8192
#define DDIM  64

// One wave32 per row: lane loads float2 (64 elems / 32 lanes), xor-reduce,
// lane 0 writes 1/sqrt(sum of squares).
__global__ void __launch_bounds__(256)
rownorm_inv_kernel(const float* __restrict__ X, float* __restrict__ inv, int rows)
{
    const int lane = threadIdx.x & 31;
    const int wave = threadIdx.x >> 5;
    const int row  = blockIdx.x * 8 + wave;
    if (row >= rows) return;

    v2f x = *(const v2f*)(X + (size_t)row * DDIM + lane * 2);
    float s = x.x * x.x + x.y * x.y;
#pragma unroll
    for (int off = 16; off > 0; off >>= 1)
        s += __shfl_xor(s, off, 32);

    if (lane == 0)
        inv[row] = 1.0f / sqrtf(s);
}

// Each wave computes a 2x2 grid of 16x16 WMMA tiles (32x32 output) with
// full-precision V_WMMA_F32_16X16X4_F32. 8 waves/block arranged 2(N) x 4(M)
// cover a 64x128 macro-tile, so A tiles are L1-shared by 4 waves and B tiles
// by 2 waves.
//
// Fragment layouts (ISA 7.12.2, f32):
//   A 16x4:  lane holds row M=lane%16, K pair = k + (lane>>4)*2 + {0,1}
//   B 4x16:  Bmat[k][n] = B[n][k] -> identical addressing on row-major B
//   C/D:     lane -> col lane%16; VGPR v -> row v + (lane>>4)*8
__global__ void __launch_bounds__(256)
cosine_wmma_f32_kernel(const float* __restrict__ A, const float* __restrict__ B,
                       const float* __restrict__ invA, const float* __restrict__ invB,
                       float* __restrict__ out)
{
    const int lane = threadIdx.x & 31;
    const int wave = threadIdx.x >> 5;
    const int wn   = wave >> 2;           // 0..1 : N direction within block
    const int wm   = wave & 3;            // 0..3 : M direction within block

    const int tN0 = blockIdx.y * 4 + wn * 2;   // first of 2 N tiles
    const int tM0 = blockIdx.x * 8 + wm * 2;   // first of 2 M tiles

    const int r16  = lane & 15;
    const int hi   = lane >> 4;
    const int koff = hi * 2;

    const float* aP0 = A + ((size_t)(tN0 + 0) * 16 + r16) * DDIM + koff;
    const float* aP1 = A + ((size_t)(tN0 + 1) * 16 + r16) * DDIM + koff;
    const float* bP0 = B + ((size_t)(tM0 + 0) * 16 + r16) * DDIM + koff;
    const float* bP1 = B + ((size_t)(tM0 + 1) * 16 + r16) * DDIM + koff;

    v8f acc00 = {0.f, 0.f, 0.f, 0.f, 0.f, 0.f, 0.f, 0.f};
    v8f acc01 = acc00, acc10 = acc00, acc11 = acc00;

#pragma unroll
    for (int k = 0; k < DDIM; k += 4) {
        v2f a0 = *(const v2f*)(aP0 + k);
        v2f a1 = *(const v2f*)(aP1 + k);
        v2f b0 = *(const v2f*)(bP0 + k);
        v2f b1 = *(const v2f*)(bP1 + k);
        // (neg_a, A, neg_b, B, c_mod, C, reuse_a, reuse_b)
        acc00 = __builtin_amdgcn_wmma_f32_16x16x4_f32(false, a0, false, b0, (short)0, acc00, false, false);
        acc01 = __builtin_amdgcn_wmma_f32_16x16x4_f32(false, a0, false, b1, (short)0, acc01, false, false);
        acc10 = __builtin_amdgcn_wmma_f32_16x16x4_f32(false, a1, false, b0, (short)0, acc10, false, false);
        acc11 = __builtin_amdgcn_wmma_f32_16x16x4_f32(false, a1, false, b1, (short)0, acc11, false, false);
    }

    const int   col0 = tM0 * 16 + r16;
    const int   col1 = col0 + 16;
    const float ib0  = invB[col0];
    const float ib1  = invB[col1];

#pragma unroll
    for (int v = 0; v < 8; ++v) {
        const int row0 = tN0 * 16 + hi * 8 + v;
        const int row1 = row0 + 16;
        const float ia0 = invA[row0];
        const float ia1 = invA[row1];
        __builtin_nontemporal_store(acc00[v] * ia0 * ib0, out + (size_t)row0 * MROWS + col0);
        __builtin_nontemporal_store(acc01[v] * ia0 * ib1, out + (size_t)row0 * MROWS + col1);
        __builtin_nontemporal_store(acc10[v] * ia1 * ib0, out + (size_t)row1 * MROWS + col0);
        __builtin_nontemporal_store(acc11[v] * ia1 * ib1, out + (size_t)row1 * MROWS + col1);
    }
}

extern "C" void kernel_launch(void* const* d_in, const int* in_sizes, int n_in,
                              void* d_out, int out_size, void* d_ws, size_t ws_size,
                              hipStream_t stream)
{
    const float* a = (const float*)d_in[0];   // [8192, 64] f32
    const float* b = (const float*)d_in[1];   // [8192, 64] f32
    float* out  = (float*)d_out;              // [8192, 8192] f32
    float* invA = (float*)d_ws;               // 8192 floats
    float* invB = invA + NROWS;               // 8192 floats

    // Inverse row norms (8 waves per 256-thread block -> 8 rows/block).
    rownorm_inv_kernel<<<NROWS / 8, 256, 0, stream>>>(a, invA, NROWS);
    rownorm_inv_kernel<<<MROWS / 8, 256, 0, stream>>>(b, invB, MROWS);

    // Block covers 64 rows x 128 cols; grid = (8192/128, 8192/64) = (64, 128).
    dim3 grid(MROWS / 128, NROWS / 64);
    cosine_wmma_f32_kernel<<<grid, 256, 0, stream>>>(a, b, invA, invB, out);
}